// GATLayer_62285615727483
// MI455X (gfx1250) — compile-verified
//
#include <hip/hip_runtime.h>
#include <hip/hip_bf16.h>
#include <stdint.h>

#define IN_F  128
#define OUT_F 64

typedef float v2f __attribute__((ext_vector_type(2)));
typedef float v8f __attribute__((ext_vector_type(8)));

// ---------------------------------------------------------------------------
// Helpers
// ---------------------------------------------------------------------------
__device__ inline void atomicMaxFloat(float* addr, float v) {
  // Order-preserving trick: non-negative floats compare as signed ints,
  // negative floats compare (reversed) as unsigned ints.
  if (v >= 0.0f) atomicMax((int*)addr, __float_as_int(v));
  else           atomicMin((unsigned int*)addr, __float_as_uint(v));
}

__device__ inline float waveReduceMax(float v) {
#pragma unroll
  for (int off = 16; off > 0; off >>= 1)
    v = fmaxf(v, __shfl_xor(v, off, 32));
  return v;
}

__device__ inline float waveReduceSum(float v) {
#pragma unroll
  for (int off = 16; off > 0; off >>= 1)
    v += __shfl_xor(v, off, 32);
  return v;
}

// ---------------------------------------------------------------------------
// Kernel 0: zero output, init softmax reduction scalars
// ---------------------------------------------------------------------------
__global__ void k_init(float* __restrict__ out, int out_n,
                       float* __restrict__ gmax, float* __restrict__ gsum) {
  int i = blockIdx.x * blockDim.x + threadIdx.x;
  if (i < out_n) out[i] = 0.0f;
  if (i == 0) { *gmax = -3.402823466e38f; *gsum = 0.0f; }
}

// ---------------------------------------------------------------------------
// Kernel 1: h = x @ W^T + b via V_WMMA_F32_16X16X4_F32
// One wave32 per 16x16 output tile. A = x tile (16x4, MxK), B = W^T tile
// (4x16, KxN). fp32 A/B fragments are 2 VGPRs/lane (v2f), C/D is v8f.
//   A layout: lanes 0-15 -> M=lane, {K=k+0,k+1}; lanes 16-31 -> M=lane-16, {K=k+2,k+3}
//   B layout: lanes 0-15 -> N=lane, {K=k+0,k+1}; lanes 16-31 -> N=lane-16, {K=k+2,k+3}
//   C layout: vgpr j, lanes 0-15 -> (M=j, N=lane); lanes 16-31 -> (M=j+8, N=lane-16)
// ---------------------------------------------------------------------------
__global__ void k_gemm_wmma(const float* __restrict__ x,     // [N,128]
                            const float* __restrict__ W,     // [64,128]
                            const float* __restrict__ Wb,    // [64]
                            float* __restrict__ h,           // [N,64]
                            int n_rows, int tiles_m) {
  const int gwave  = (blockIdx.x * blockDim.x + threadIdx.x) >> 5;
  const int lane   = threadIdx.x & 31;
  const int tile_m = gwave >> 2;            // 4 N-tiles (64/16)
  const int tile_n = gwave & 3;
  if (tile_m >= tiles_m) return;            // uniform across the wave

  const int half  = lane >> 4;              // 0: K=k..k+1, 1: K=k+2..k+3
  const int l16   = lane & 15;
  const int khalf = half * 2;

  int m = tile_m * 16 + l16;
  if (m >= n_rows) m = n_rows - 1;          // clamp loads, keep EXEC all-1s
  const int n = tile_n * 16 + l16;

  const float* __restrict__ xrow = x + (size_t)m * IN_F;
  const float* __restrict__ wrow = W + (size_t)n * IN_F;

  v8f c = {};
#pragma unroll
  for (int k = 0; k < IN_F; k += 4) {
    v2f a = *(const v2f*)(xrow + k + khalf);   // 8B-aligned (k+khalf even)
    v2f b = *(const v2f*)(wrow + k + khalf);
    c = __builtin_amdgcn_wmma_f32_16x16x4_f32(
        /*neg_a=*/false, a, /*neg_b=*/false, b,
        /*c_mod=*/(short)0, c, /*reuse_a=*/false, /*reuse_b=*/false);
  }

  const float bias  = Wb[n];
  const int   rbase = tile_m * 16 + half * 8;
#pragma unroll
  for (int j = 0; j < 8; ++j) {
    const int r = rbase + j;
    if (r < n_rows) h[(size_t)r * OUT_F + n] = c[j] + bias;
  }
}

// ---------------------------------------------------------------------------
// Kernel 2: per-node attention scalars  sr[n]=h[n].a_r  sc[n]=h[n].a_c
// ---------------------------------------------------------------------------
__global__ void k_node_scalars(const float* __restrict__ h,
                               const float* __restrict__ a_w,  // [128]
                               float* __restrict__ sr, float* __restrict__ sc,
                               int n_rows) {
  const int n = blockIdx.x * blockDim.x + threadIdx.x;
  if (n >= n_rows) return;
  const float4* __restrict__ h4 = (const float4*)(h + (size_t)n * OUT_F);
  const float4* __restrict__ ar = (const float4*)(a_w);
  const float4* __restrict__ ac = (const float4*)(a_w + OUT_F);
  float r = 0.0f, cacc = 0.0f;
#pragma unroll
  for (int q = 0; q < OUT_F / 4; ++q) {
    const float4 v = h4[q];
    const float4 wr = ar[q], wc = ac[q];
    r    = fmaf(v.x, wr.x, fmaf(v.y, wr.y, fmaf(v.z, wr.z, fmaf(v.w, wr.w, r))));
    cacc = fmaf(v.x, wc.x, fmaf(v.y, wc.y, fmaf(v.z, wc.z, fmaf(v.w, wc.w, cacc))));
  }
  sr[n] = r;
  sc[n] = cacc;
}

// ---------------------------------------------------------------------------
// Kernel 3: per-edge score = leaky_relu(sr[row]+sc[col]+a_b), global max
// ---------------------------------------------------------------------------
__global__ void k_edge_score_max(const long long* __restrict__ ei,   // [2,E] int64
                                 const float* __restrict__ sr,
                                 const float* __restrict__ sc,
                                 const float* __restrict__ a_b,
                                 float* __restrict__ score,
                                 float* __restrict__ gmax, int E) {
  const int e = blockIdx.x * blockDim.x + threadIdx.x;
  float s = -3.402823466e38f;
  if (e < E) {
    const int r = (int)ei[e];
    const int c = (int)ei[(size_t)E + e];
    const float v = sr[r] + sc[c] + a_b[0];
    s = (v >= 0.0f) ? v : 0.2f * v;       // leaky_relu(0.2)
    score[e] = s;
  }
  __shared__ float red[8];
  s = waveReduceMax(s);
  if ((threadIdx.x & 31) == 0) red[threadIdx.x >> 5] = s;
  __syncthreads();
  if (threadIdx.x < 8) {
    float w = red[threadIdx.x];
#pragma unroll
    for (int off = 4; off > 0; off >>= 1) w = fmaxf(w, __shfl_xor(w, off, 8));
    if (threadIdx.x == 0) atomicMaxFloat(gmax, w);
  }
}

// ---------------------------------------------------------------------------
// Kernel 4: w_e = exp(score_e - max), global sum; overwrite score with w_e
// ---------------------------------------------------------------------------
__global__ void k_edge_exp_sum(float* __restrict__ score,
                               const float* __restrict__ gmax,
                               float* __restrict__ gsum, int E) {
  const int e = blockIdx.x * blockDim.x + threadIdx.x;
  float w = 0.0f;
  if (e < E) {
    w = __expf(score[e] - *gmax);
    score[e] = w;
  }
  __shared__ float red[8];
  w = waveReduceSum(w);
  if ((threadIdx.x & 31) == 0) red[threadIdx.x >> 5] = w;
  __syncthreads();
  if (threadIdx.x < 8) {
    float t = red[threadIdx.x];
#pragma unroll
    for (int off = 4; off > 0; off >>= 1) t += __shfl_xor(t, off, 8);
    if (threadIdx.x == 0) atomicAdd(gsum, t);
  }
}

// ---------------------------------------------------------------------------
// Kernel 5: out[row] += (w_e / sum) * h[col]; 16 threads/edge, 4 floats each.
// h and out are L2-resident (25.6 MB each vs 192 MB L2) so the random
// gathers and f32 atomics mostly resolve in L2.
// ---------------------------------------------------------------------------
__global__ void k_edge_scatter(const long long* __restrict__ ei,
                               const float* __restrict__ score,   // = w_e
                               const float* __restrict__ gsum,
                               const float* __restrict__ h,
                               float* __restrict__ out, int E) {
  const long long t = (long long)blockIdx.x * blockDim.x + threadIdx.x;
  const int e = (int)(t >> 4);
  if (e >= E) return;
  const int j = ((int)t & 15) * 4;
  const float alpha = score[e] * (1.0f / *gsum);
  const int r = (int)ei[e];
  const int c = (int)ei[(size_t)E + e];
  const float4 hv = *(const float4*)(h + (size_t)c * OUT_F + j);
  float* o = out + (size_t)r * OUT_F + j;
  atomicAdd(o + 0, alpha * hv.x);
  atomicAdd(o + 1, alpha * hv.y);
  atomicAdd(o + 2, alpha * hv.z);
  atomicAdd(o + 3, alpha * hv.w);
}

// ---------------------------------------------------------------------------
// Launch
// ---------------------------------------------------------------------------
extern "C" void kernel_launch(void* const* d_in, const int* in_sizes, int n_in,
                              void* d_out, int out_size, void* d_ws, size_t ws_size,
                              hipStream_t stream) {
  const float*     x   = (const float*)d_in[0];       // [N,128]
  const long long* ei  = (const long long*)d_in[1];   // [2,E] int64
  const float*     W_w = (const float*)d_in[2];       // [64,128]
  const float*     W_b = (const float*)d_in[3];       // [64]
  const float*     a_w = (const float*)d_in[4];       // [1,128]
  const float*     a_b = (const float*)d_in[5];       // [1]
  float* out = (float*)d_out;

  const int N = in_sizes[0] / IN_F;
  const int E = in_sizes[1] / 2;

  // Workspace layout (floats)
  float* ws    = (float*)d_ws;
  float* h     = ws;                                  // N*64
  float* sr    = h  + (size_t)N * OUT_F;              // N
  float* sc    = sr + N;                              // N
  float* score = sc + N;                              // E
  float* gmax  = score + E;                           // 1
  float* gsum  = gmax + 1;                            // 1

  // 0) init out + reduction scalars
  {
    int n = out_size > 1 ? out_size : 1;
    k_init<<<(n + 255) / 256, 256, 0, stream>>>(out, out_size, gmax, gsum);
  }
  // 1) WMMA GEMM: h = x W^T + b
  {
    const int tiles_m = (N + 15) / 16;
    const int waves   = tiles_m * 4;                  // 4 N-tiles
    k_gemm_wmma<<<(waves + 7) / 8, 256, 0, stream>>>(x, W_w, W_b, h, N, tiles_m);
  }
  // 2) per-node attention scalars
  k_node_scalars<<<(N + 255) / 256, 256, 0, stream>>>(h, a_w, sr, sc, N);
  // 3) edge scores + global max
  k_edge_score_max<<<(E + 255) / 256, 256, 0, stream>>>(ei, sr, sc, a_b, score, gmax, E);
  // 4) exp + global sum
  k_edge_exp_sum<<<(E + 255) / 256, 256, 0, stream>>>(score, gmax, gsum, E);
  // 5) scatter-accumulate
  {
    const long long threads = (long long)E * 16;
    k_edge_scatter<<<(unsigned)((threads + 255) / 256), 256, 0, stream>>>(
        ei, score, gsum, h, out, E);
  }
}